// HSI_MambaVisionLayer_62663572848661
// MI455X (gfx1250) — compile-verified
//
#include <hip/hip_runtime.h>
#include <hip/hip_bf16.h>

// MI455X (gfx1250) implementation. All GEMM-shaped math uses
// V_WMMA_F32_16X16X4_F32 (fp32 in, fp32 accumulate) -- inputs are fp32 and the
// problem (~40 GFLOP, ~100s of MB of intermediates at 23.3 TB/s) is
// bandwidth/latency bound, so full-precision WMMA is the right pick over bf16.
// GEMM waves are register-blocked 16x64 (4 C tiles share one A fragment) and
// manually double-buffered so fragment loads overlap the WMMA issue.

typedef __attribute__((ext_vector_type(2))) float v2f;
typedef __attribute__((ext_vector_type(8))) float v8f;

__device__ __forceinline__ v8f wmma4(v2f a, v2f b, v8f c) {
    // D = A(16x4) * B(4x16) + C(16x16), fp32
    return __builtin_amdgcn_wmma_f32_16x16x4_f32(false, a, false, b, (short)0, c,
                                                 false, false);
}

enum { EPI_NONE = 0, EPI_BIAS = 1, EPI_BIAS_GELU = 2, EPI_RES = 3, EPI_BIAS_RES = 4 };

// ---------------------------------------------------------------- window ops
__global__ void win_part_kernel(const float* __restrict__ x, float* __restrict__ t) {
    int i = blockIdx.x * blockDim.x + threadIdx.x;      // 64*512*128
    if (i >= 64 * 512 * 128) return;
    int c = i & 127;
    int n = (i >> 7) & 511;
    int win = i >> 16;
    int b = win >> 5; int rem = win & 31;
    int dw = rem >> 4; int hw = (rem >> 2) & 3; int ww = rem & 3;
    int ld = n >> 6; int lh = (n >> 3) & 7; int lw = n & 7;
    int d = dw * 8 + ld, h = hw * 8 + lh, w = ww * 8 + lw;
    t[i] = x[(((size_t)(b * 128 + c) * 16 + d) * 32 + h) * 32 + w];
}

__global__ void win_rev_kernel(const float* __restrict__ t, float* __restrict__ x) {
    int i = blockIdx.x * blockDim.x + threadIdx.x;      // 2*128*16*32*32
    if (i >= 2 * 128 * 16 * 32 * 32) return;
    int w = i & 31; int h = (i >> 5) & 31; int d = (i >> 10) & 15;
    int c = (i >> 14) & 127; int b = i >> 21;
    int win = b * 32 + (d >> 3) * 16 + (h >> 3) * 4 + (w >> 3);
    int n = (d & 7) * 64 + (h & 7) * 8 + (w & 7);
    x[i] = t[((size_t)win * 512 + n) * 128 + c];
}

// ---------------------------------------------------------------- layernorm
__global__ void ln_kernel(const float* __restrict__ in, const float* __restrict__ g,
                          const float* __restrict__ bb, float* __restrict__ out, int T) {
    int wave = (blockIdx.x * blockDim.x + threadIdx.x) >> 5;
    int lane = threadIdx.x & 31;
    if (wave >= T) return;
    const float* row = in + (size_t)wave * 128;
    float v0 = row[lane], v1 = row[lane + 32], v2 = row[lane + 64], v3 = row[lane + 96];
    float s = v0 + v1 + v2 + v3;
    for (int m = 1; m < 32; m <<= 1) s += __shfl_xor(s, m, 32);
    float mu = s * (1.f / 128.f);
    float d0 = v0 - mu, d1 = v1 - mu, d2 = v2 - mu, d3 = v3 - mu;
    float q = d0 * d0 + d1 * d1 + d2 * d2 + d3 * d3;
    for (int m = 1; m < 32; m <<= 1) q += __shfl_xor(q, m, 32);
    float inv = rsqrtf(q * (1.f / 128.f) + 1e-5f);
    float* orow = out + (size_t)wave * 128;
    orow[lane]      = d0 * inv * g[lane]      + bb[lane];
    orow[lane + 32] = d1 * inv * g[lane + 32] + bb[lane + 32];
    orow[lane + 64] = d2 * inv * g[lane + 64] + bb[lane + 64];
    orow[lane + 96] = d3 * inv * g[lane + 96] + bb[lane + 96];
}

// ---------------------------------------------------------------- WMMA GEMM
// out[T,N] = A[T,K] * W[N,K]^T (+bias)(+act)(+residual).
// One wave per 16x64 output tile: 4 accumulators share each A fragment.
// Double-buffered K loop: next step's 5 fragments load while current 4 WMMAs issue.
__global__ void gemm_wmma_kernel(const float* __restrict__ A, const float* __restrict__ W,
                                 const float* __restrict__ bias, const float* __restrict__ res,
                                 float* __restrict__ out, int T, int N, int K, int epi) {
    int wave = (blockIdx.x * blockDim.x + threadIdx.x) >> 5;
    int lane = threadIdx.x & 31;
    int nt = N >> 6;                              // 64-wide column groups
    int tiles = (T >> 4) * nt;
    if (wave >= tiles) return;
    int rowBase = (wave / nt) << 4;
    int colBase = (wave % nt) << 6;
    int hx = lane >> 4, l16 = lane & 15;
    const float* arow = A + (size_t)(rowBase + l16) * K + 2 * hx;
    const float* wrow = W + (size_t)(colBase + l16) * K + 2 * hx;
    size_t ws1 = (size_t)16 * K, ws2 = 2 * ws1, ws3 = 3 * ws1;
    v8f acc0 = {}, acc1 = {}, acc2 = {}, acc3 = {};
    v2f a  = *(const v2f*)(arow);
    v2f b0 = *(const v2f*)(wrow);
    v2f b1 = *(const v2f*)(wrow + ws1);
    v2f b2 = *(const v2f*)(wrow + ws2);
    v2f b3 = *(const v2f*)(wrow + ws3);
    for (int k = 4; k < K; k += 4) {
        v2f an  = *(const v2f*)(arow + k);
        v2f bn0 = *(const v2f*)(wrow + k);
        v2f bn1 = *(const v2f*)(wrow + ws1 + k);
        v2f bn2 = *(const v2f*)(wrow + ws2 + k);
        v2f bn3 = *(const v2f*)(wrow + ws3 + k);
        acc0 = wmma4(a, b0, acc0);
        acc1 = wmma4(a, b1, acc1);
        acc2 = wmma4(a, b2, acc2);
        acc3 = wmma4(a, b3, acc3);
        a = an; b0 = bn0; b1 = bn1; b2 = bn2; b3 = bn3;
    }
    acc0 = wmma4(a, b0, acc0);
    acc1 = wmma4(a, b1, acc1);
    acc2 = wmma4(a, b2, acc2);
    acc3 = wmma4(a, b3, acc3);

    bool hasBias = (epi == EPI_BIAS || epi == EPI_BIAS_GELU || epi == EPI_BIAS_RES);
    bool hasRes  = (epi == EPI_RES || epi == EPI_BIAS_RES);
    v8f accs[4] = {acc0, acc1, acc2, acc3};
#pragma unroll
    for (int j = 0; j < 4; ++j) {
        int col = colBase + j * 16 + l16;
        float bv = hasBias ? bias[col] : 0.f;
#pragma unroll
        for (int r = 0; r < 8; ++r) {
            int row = rowBase + r + 8 * hx;
            size_t idx = (size_t)row * N + col;
            float v = accs[j][r] + bv;
            if (epi == EPI_BIAS_GELU)
                v = 0.5f * v * (1.f + erff(v * 0.70710678118654752f));
            if (hasRes) v += res[idx];
            out[idx] = v;
        }
    }
}

// ------------------------------------------------ mamba conv + SSM (serial L)
__global__ void mamba_ssm_kernel(const float* __restrict__ xz, const float* __restrict__ cw,
                                 const float* __restrict__ cb, const float* __restrict__ Bp,
                                 const float* __restrict__ Cp, const float* __restrict__ Dp,
                                 float* __restrict__ u) {
    int i = blockIdx.x * blockDim.x + threadIdx.x;   // 64 windows * 128 channels
    if (i >= 64 * 128) return;
    int c = i & 127, win = i >> 7;
    float w0 = cw[c * 3 + 0], w1 = cw[c * 3 + 1], w2 = cw[c * 3 + 2], b = cb[c];
    float bc = 0.f;
#pragma unroll
    for (int s = 0; s < 8; ++s) bc += Bp[c * 8 + s] * Cp[c * 8 + s];
    float Dv = Dp[c];
    float xm1 = 0.f, xm2 = 0.f, run = 0.f;
    const float* base = xz + (size_t)win * 512 * 256;
    float* ub = u + (size_t)win * 512 * 128;
    for (int l = 0; l < 512; ++l) {
        float xc = base[l * 256 + c];
        float z  = base[l * 256 + 128 + c];
        float conv = w0 * xm2 + w1 * xm1 + w2 * xc + b;
        xm2 = xm1; xm1 = xc;
        float sx = conv / (1.f + expf(-conv));   // silu
        run += sx;
        float y = run * bc + sx * Dv;
        float sz = z / (1.f + expf(-z));
        ub[l * 128 + c] = y * sz;
    }
}

// ---------------------------------------------------------------- attention
// One block (4 waves) per (window, head, 16-query tile). S(16x512) in LDS.
__global__ void attn_kernel(const float* __restrict__ qkv, float* __restrict__ out) {
    __shared__ float S[16][512];
    __shared__ float Op[4][16][32];
    int tid = threadIdx.x;
    int waveId = tid >> 5, lane = tid & 31, hx = lane >> 4, l16 = lane & 15;
    int blk = blockIdx.x;                  // 64*4*32 blocks
    int win = blk >> 7; int rem = blk & 127; int head = rem >> 5; int qt = rem & 31;
    int qBase = qt << 4;
    const float* qptr = qkv + ((size_t)win * 512 + qBase + l16) * 384 + head * 32 + 2 * hx;
    // phase 1: S = (Q K^T) * scale
    for (int kt = waveId; kt < 32; kt += 4) {
        v8f acc = {};
        const float* kptr =
            qkv + ((size_t)win * 512 + kt * 16 + l16) * 384 + 128 + head * 32 + 2 * hx;
#pragma unroll
        for (int kk = 0; kk < 32; kk += 4) {
            v2f a = *(const v2f*)(qptr + kk);
            v2f b = *(const v2f*)(kptr + kk);
            acc = wmma4(a, b, acc);
        }
#pragma unroll
        for (int r = 0; r < 8; ++r)
            S[r + 8 * hx][kt * 16 + l16] = acc[r] * 0.17677669529663687f;
    }
    __syncthreads();
    // phase 2: softmax rows (16 rows x 8 threads each)
    {
        int row = tid >> 3, sub = tid & 7;
        float m = -3.4e38f;
        for (int c2 = sub; c2 < 512; c2 += 8) m = fmaxf(m, S[row][c2]);
        for (int s = 1; s < 8; s <<= 1) m = fmaxf(m, __shfl_xor(m, s, 32));
        float sum = 0.f;
        for (int c2 = sub; c2 < 512; c2 += 8) {
            float p = expf(S[row][c2] - m);
            S[row][c2] = p;
            sum += p;
        }
        for (int s = 1; s < 8; s <<= 1) sum += __shfl_xor(sum, s, 32);
        float inv = 1.f / sum;
        for (int c2 = sub; c2 < 512; c2 += 8) S[row][c2] *= inv;
    }
    __syncthreads();
    // phase 3: O = P V, keys split across 4 waves (128 each)
    int kb = waveId * 128;
    v8f acc0 = {}, acc1 = {};
    for (int kk = 0; kk < 128; kk += 4) {
        v2f a = *(const v2f*)(&S[l16][kb + kk + 2 * hx]);
        const float* vb =
            qkv + ((size_t)win * 512 + kb + kk + 2 * hx) * 384 + 256 + head * 32;
        v2f b0, b1;
        b0.x = vb[l16];          b0.y = vb[384 + l16];
        b1.x = vb[16 + l16];     b1.y = vb[384 + 16 + l16];
        acc0 = wmma4(a, b0, acc0);
        acc1 = wmma4(a, b1, acc1);
    }
#pragma unroll
    for (int r = 0; r < 8; ++r) {
        Op[waveId][r + 8 * hx][l16]      = acc0[r];
        Op[waveId][r + 8 * hx][16 + l16] = acc1[r];
    }
    __syncthreads();
    for (int idx = tid; idx < 512; idx += 128) {
        int m2 = idx >> 5, n2 = idx & 31;
        float s = Op[0][m2][n2] + Op[1][m2][n2] + Op[2][m2][n2] + Op[3][m2][n2];
        out[((size_t)win * 512 + qBase + m2) * 128 + head * 32 + n2] = s;
    }
}

// --------------------------------------------- conv3d 128->256 k3 s2 p1 (implicit GEMM)
__global__ void conv_down_kernel(const float* __restrict__ x, const float* __restrict__ w,
                                 float* __restrict__ out) {
    int wave = (blockIdx.x * blockDim.x + threadIdx.x) >> 5;
    int lane = threadIdx.x & 31, hx = lane >> 4, l16 = lane & 15;
    if (wave >= 4096) return;                 // 256 m-tiles * 16 n-tiles
    int nTile = wave & 15, mTile = wave >> 4;
    int nb = mTile >> 7; int rem = mTile & 127; int od = rem >> 4; int oh = rem & 15;
    int ocBase = nTile << 4;
    int ow = l16;                             // M index = output width position
    v8f acc = {};
    for (int kd = 0; kd < 3; ++kd) {
        int id = 2 * od - 1 + kd;
        if (id < 0 || id >= 16) continue;
        for (int kh = 0; kh < 3; ++kh) {
            int ih = 2 * oh - 1 + kh;
            if (ih < 0 || ih >= 32) continue;
            for (int kw = 0; kw < 3; ++kw) {
                int iw = 2 * ow - 1 + kw;
                bool valid = (iw >= 0) && (iw < 32);
                int tap = (kd * 3 + kh) * 3 + kw;
                const float* xb =
                    x + (((size_t)nb * 128) * 16 + id) * 1024 + ih * 32 + iw;
                const float* wb = w + (size_t)(ocBase + l16) * 128 * 27 + tap;
                for (int kk = 0; kk < 128; kk += 4) {
                    int ic = kk + 2 * hx;
                    v2f a, b;
                    a.x = valid ? xb[(size_t)ic * 16384] : 0.f;
                    a.y = valid ? xb[(size_t)(ic + 1) * 16384] : 0.f;
                    b.x = wb[ic * 27];
                    b.y = wb[(ic + 1) * 27];
                    acc = wmma4(a, b, acc);
                }
            }
        }
    }
#pragma unroll
    for (int r = 0; r < 8; ++r) {
        int oww = r + 8 * hx;
        int oc = ocBase + l16;
        out[(((size_t)nb * 256 + oc) * 8 + od) * 256 + oh * 16 + oww] = acc[r];
    }
}

// ---------------------------------------------------------------- launcher
extern "C" void kernel_launch(void* const* d_in, const int* in_sizes, int n_in,
                              void* d_out, int out_size, void* d_ws, size_t ws_size,
                              hipStream_t stream) {
    const float* x          = (const float*)d_in[0];
    const float* ln1_0_g    = (const float*)d_in[1];
    const float* ln1_0_b    = (const float*)d_in[2];
    const float* mamba_in_w = (const float*)d_in[3];
    const float* mamba_cw   = (const float*)d_in[4];
    const float* mamba_cb   = (const float*)d_in[5];
    const float* mamba_B    = (const float*)d_in[6];
    const float* mamba_C    = (const float*)d_in[7];
    const float* mamba_D    = (const float*)d_in[8];
    const float* mamba_ow   = (const float*)d_in[9];
    const float* ln2_0_g    = (const float*)d_in[10];
    const float* ln2_0_b    = (const float*)d_in[11];
    const float* mlp0_w1    = (const float*)d_in[12];
    const float* mlp0_b1    = (const float*)d_in[13];
    const float* mlp0_w2    = (const float*)d_in[14];
    const float* mlp0_b2    = (const float*)d_in[15];
    const float* ln1_1_g    = (const float*)d_in[16];
    const float* ln1_1_b    = (const float*)d_in[17];
    const float* qkv_w      = (const float*)d_in[18];
    const float* qkv_b      = (const float*)d_in[19];
    const float* proj_w     = (const float*)d_in[20];
    const float* proj_b     = (const float*)d_in[21];
    const float* ln2_1_g    = (const float*)d_in[22];
    const float* ln2_1_b    = (const float*)d_in[23];
    const float* mlp1_w1    = (const float*)d_in[24];
    const float* mlp1_b1    = (const float*)d_in[25];
    const float* mlp1_w2    = (const float*)d_in[26];
    const float* mlp1_b2    = (const float*)d_in[27];
    const float* down_w     = (const float*)d_in[28];

    const int T = 64 * 512;                     // 32768 tokens
    float* t   = (float*)d_ws;                  // (T,128)  4.19M floats
    float* tn  = t + 4194304;                   // (T,128)  normed / x2 scratch
    float* big = tn + 4194304;                  // 16.78M floats: xz / h / qkv(+attnout)
    float* uM  = big + 8388608;                 // mamba gate-out (after xz region)
    float* uA  = big + 12582912;                // attention out (after qkv region)

    // partition
    win_part_kernel<<<16384, 256, 0, stream>>>(x, t);

    // block 0: mamba   (gemm waves = (T/16)*(N/64), 4 waves/block of 128 thr)
    ln_kernel<<<4096, 256, 0, stream>>>(t, ln1_0_g, ln1_0_b, tn, T);
    gemm_wmma_kernel<<<2048, 128, 0, stream>>>(tn, mamba_in_w, nullptr, nullptr,
                                               big, T, 256, 128, EPI_NONE);
    mamba_ssm_kernel<<<64, 128, 0, stream>>>(big, mamba_cw, mamba_cb,
                                             mamba_B, mamba_C, mamba_D, uM);
    gemm_wmma_kernel<<<1024, 128, 0, stream>>>(uM, mamba_ow, nullptr, t,
                                               t, T, 128, 128, EPI_RES);
    // mlp0
    ln_kernel<<<4096, 256, 0, stream>>>(t, ln2_0_g, ln2_0_b, tn, T);
    gemm_wmma_kernel<<<4096, 128, 0, stream>>>(tn, mlp0_w1, mlp0_b1, nullptr,
                                               big, T, 512, 128, EPI_BIAS_GELU);
    gemm_wmma_kernel<<<1024, 128, 0, stream>>>(big, mlp0_w2, mlp0_b2, t,
                                               t, T, 128, 512, EPI_BIAS_RES);
    // block 1: attention
    ln_kernel<<<4096, 256, 0, stream>>>(t, ln1_1_g, ln1_1_b, tn, T);
    gemm_wmma_kernel<<<3072, 128, 0, stream>>>(tn, qkv_w, qkv_b, nullptr,
                                               big, T, 384, 128, EPI_BIAS);
    attn_kernel<<<8192, 128, 0, stream>>>(big, uA);
    gemm_wmma_kernel<<<1024, 128, 0, stream>>>(uA, proj_w, proj_b, t,
                                               t, T, 128, 128, EPI_BIAS_RES);
    // mlp1
    ln_kernel<<<4096, 256, 0, stream>>>(t, ln2_1_g, ln2_1_b, tn, T);
    gemm_wmma_kernel<<<4096, 128, 0, stream>>>(tn, mlp1_w1, mlp1_b1, nullptr,
                                               big, T, 512, 128, EPI_BIAS_GELU);
    gemm_wmma_kernel<<<1024, 128, 0, stream>>>(big, mlp1_w2, mlp1_b2, t,
                                               t, T, 128, 512, EPI_BIAS_RES);
    // reverse + downsample
    win_rev_kernel<<<16384, 256, 0, stream>>>(t, tn);
    conv_down_kernel<<<1024, 128, 0, stream>>>(tn, down_w, (float*)d_out);
}